// MultiHeadAttention3_11493332484225
// MI455X (gfx1250) — compile-verified
//
#include <hip/hip_runtime.h>
#include <hip/hip_bf16.h>
#include <math.h>

// MHA forward for B=2, S=2048, D=256, H=8, Dh=32 (all f32) using
// V_WMMA_F32_16X16X4_F32 on gfx1250 (wave32).

typedef __attribute__((ext_vector_type(2))) float v2f;
typedef __attribute__((ext_vector_type(8))) float v8f;

#define BATCH 2
#define SEQ   2048
#define DMODEL 256
#define NHEAD 8
#define DHEAD 32
#define SCALE 0.17677669529663687f  /* 1/sqrt(32) */

// -------------------------------------------------------------------------
// Phase 1: fused QKV projections.  y = x @ W^T + b  for W in {wq,wk,wv}.
// Q,V stored [B,H,S,Dh]; K stored transposed [B,H,Dh,S] for the score GEMM.
// Register-blocked: one wave produces a 16x64 tile (4 N-subtiles), so each
// A-fragment is reused by 4 WMMAs (load:WMMA ratio 1.25 instead of 2.0).
// Waves: 3 mats * (4096/16 Mtiles) * (256/64 Ngroups) = 3072.
// -------------------------------------------------------------------------
__global__ __launch_bounds__(128) void qkv_kernel(
    const float* __restrict__ x,
    const float* __restrict__ wq, const float* __restrict__ bq,
    const float* __restrict__ wk, const float* __restrict__ bk,
    const float* __restrict__ wv, const float* __restrict__ bv,
    float* __restrict__ Q, float* __restrict__ Kt, float* __restrict__ V)
{
    const int wave = (blockIdx.x * blockDim.x + threadIdx.x) >> 5;
    const int lane = threadIdx.x & 31;
    const int mat  = wave >> 10;            // 0=Q 1=K 2=V (1024 waves each)
    const int rem  = wave & 1023;
    const int m0   = (rem >> 2) << 4;       // token-tile base (0..4080)
    const int n0   = (rem & 3) << 6;        // feature-group base (0,64,128,192)

    const float* w;  const float* bias;
    if (mat == 0)      { w = wq; bias = bq; }
    else if (mat == 1) { w = wk; bias = bk; }
    else               { w = wv; bias = bv; }

    const int mn = lane & 15;               // A: row m / B: col n
    const int kk = (lane >> 4) << 1;        // k-pair base within chunk (0 or 2)

    v8f acc[4];
    for (int j = 0; j < 4; ++j) {
        const float bb = bias[n0 + j * 16 + mn];   // bias depends on col only
        for (int r = 0; r < 8; ++r) acc[j][r] = bb;
    }

    const float* xr = x + (size_t)(m0 + mn) * DMODEL;
    const float* wr0 = w + (size_t)(n0 +  0 + mn) * DMODEL;  // B[k][n]=w[n][k]
    const float* wr1 = w + (size_t)(n0 + 16 + mn) * DMODEL;
    const float* wr2 = w + (size_t)(n0 + 32 + mn) * DMODEL;
    const float* wr3 = w + (size_t)(n0 + 48 + mn) * DMODEL;

    for (int kc = 0; kc < DMODEL / 4; ++kc) {
        const int ko = kc * 4 + kk;
        v2f a;  a.x = xr[ko];  a.y = xr[ko + 1];
        v2f b0; b0.x = wr0[ko]; b0.y = wr0[ko + 1];
        v2f b1; b1.x = wr1[ko]; b1.y = wr1[ko + 1];
        v2f b2; b2.x = wr2[ko]; b2.y = wr2[ko + 1];
        v2f b3; b3.x = wr3[ko]; b3.y = wr3[ko + 1];
        acc[0] = __builtin_amdgcn_wmma_f32_16x16x4_f32(
            false, a, false, b0, (short)0, acc[0], false, false);
        acc[1] = __builtin_amdgcn_wmma_f32_16x16x4_f32(
            false, a, false, b1, (short)0, acc[1], false, false);
        acc[2] = __builtin_amdgcn_wmma_f32_16x16x4_f32(
            false, a, false, b2, (short)0, acc[2], false, false);
        acc[3] = __builtin_amdgcn_wmma_f32_16x16x4_f32(
            false, a, false, b3, (short)0, acc[3], false, false);
    }

    // D layout: vgpr r -> row m0 + r + 8*(lane>=16); col = subtile base + mn
    const int rowb = m0 + ((lane >> 4) << 3);
    for (int j = 0; j < 4; ++j) {
        const int col = n0 + j * 16 + mn;
        const int h   = col >> 5;
        const int dh  = col & 31;
        for (int r = 0; r < 8; ++r) {
            const int mrow = rowb + r;      // global token index 0..4095
            const int b_ = mrow >> 11;      // batch
            const int s  = mrow & 2047;     // seq position
            if (mat == 1) {
                Kt[((size_t)(b_ * NHEAD + h) * DHEAD + dh) * SEQ + s] = acc[j][r];
            } else {
                float* dst = (mat == 0) ? Q : V;
                dst[((size_t)(b_ * NHEAD + h) * SEQ + s) * DHEAD + dh] = acc[j][r];
            }
        }
    }
}

// -------------------------------------------------------------------------
// Phase 2: attention. One workgroup per (b, h, 16-row q tile); 4 waves each
// owning a 512-wide k range. Full 16x2048 score block in LDS (padded stride).
// -------------------------------------------------------------------------
#define SD 2052   // 2048 + 4 pad -> rows spread across LDS banks

__global__ __launch_bounds__(128) void attn_kernel(
    const float* __restrict__ Q, const float* __restrict__ Kt,
    const float* __restrict__ V,
    float* __restrict__ ctx, float* __restrict__ attn_out)
{
    __shared__ float sc[16 * SD];        // score / prob tile, ~128 KB
    __shared__ float red[16 * 8];        // per-segment reductions
    __shared__ float rstat[16];          // row max, then 1/rowsum
    __shared__ float part[4 * 16 * 32];  // per-wave P@V partials

    const int wid  = threadIdx.x >> 5;   // wave 0..3
    const int lane = threadIdx.x & 31;
    const int blk   = blockIdx.x;        // b*1024 + h*128 + qtile
    const int qtile = blk & 127;
    const int h     = (blk >> 7) & 7;
    const int b     = blk >> 10;
    const int q0    = qtile << 4;
    const int bh    = b * NHEAD + h;

    const float* Qp  = Q  + ((size_t)bh * SEQ + q0) * DHEAD;
    const float* Ktp = Kt + (size_t)bh * DHEAD * SEQ;
    const float* Vp  = V  + (size_t)bh * SEQ * DHEAD;

    const int mn = lane & 15;
    const int kk = (lane >> 4) << 1;
    const int rowb = (lane >> 4) << 3;

    // Q A-fragments, 8 chunks of K=4 covering Dh=32 (reused for all k-blocks)
    v2f aq[8];
    for (int c = 0; c < 8; ++c) {
        aq[c].x = Qp[(size_t)mn * DHEAD + c * 4 + kk];
        aq[c].y = Qp[(size_t)mn * DHEAD + c * 4 + kk + 1];
    }

    // scores: wave wid covers k blocks [wid*32, wid*32+32)
    for (int kb = wid * 32; kb < wid * 32 + 32; ++kb) {
        const int k0 = kb << 4;
        v8f acc = {};
        for (int c = 0; c < 8; ++c) {
            v2f bf;   // B[d][k] = Kt[d][k]
            bf.x = Ktp[(size_t)(c * 4 + kk)     * SEQ + k0 + mn];
            bf.y = Ktp[(size_t)(c * 4 + kk + 1) * SEQ + k0 + mn];
            acc = __builtin_amdgcn_wmma_f32_16x16x4_f32(
                false, aq[c], false, bf, (short)0, acc, false, false);
        }
        for (int r = 0; r < 8; ++r)
            sc[(rowb + r) * SD + k0 + mn] = acc[r] * SCALE;
    }
    __syncthreads();

    // softmax: 128 threads -> row = tid/8, segment of 256 cols = tid%8
    const int row = threadIdx.x >> 3;
    const int seg = threadIdx.x & 7;
    float* srow = sc + row * SD + seg * 256;

    float mx = -3.0e38f;
    for (int i = 0; i < 256; ++i) mx = fmaxf(mx, srow[i]);
    red[row * 8 + seg] = mx;
    __syncthreads();
    if (threadIdx.x < 16) {
        float m_ = red[threadIdx.x * 8];
        for (int i = 1; i < 8; ++i) m_ = fmaxf(m_, red[threadIdx.x * 8 + i]);
        rstat[threadIdx.x] = m_;
    }
    __syncthreads();
    const float rm = rstat[row];
    float sum = 0.f;
    for (int i = 0; i < 256; ++i) {
        const float e = __expf(srow[i] - rm);
        srow[i] = e;
        sum += e;
    }
    red[row * 8 + seg] = sum;
    __syncthreads();
    if (threadIdx.x < 16) {
        float s_ = 0.f;
        for (int i = 0; i < 8; ++i) s_ += red[threadIdx.x * 8 + i];
        rstat[threadIdx.x] = 1.0f / s_;
    }
    __syncthreads();
    const float rinv = rstat[row];

    // normalize in place + accumulate head-mean attention (pre-zeroed)
    float* arow = attn_out + ((size_t)b * SEQ + q0 + row) * SEQ + seg * 256;
    for (int i = 0; i < 256; ++i) {
        const float p = srow[i] * rinv;
        srow[i] = p;
        atomicAdd(&arow[i], p * 0.125f);
    }
    __syncthreads();

    // P @ V : wave wid covers k chunks [wid*128, wid*128+128),
    // n-tiles d0..15 / d16..31; A read from LDS, B from V.
    v8f o0 = {}, o1 = {};
    for (int kc = wid * 128; kc < wid * 128 + 128; ++kc) {
        v2f ap;
        ap.x = sc[mn * SD + kc * 4 + kk];
        ap.y = sc[mn * SD + kc * 4 + kk + 1];
        v2f b0, b1;
        b0.x = Vp[(size_t)(kc * 4 + kk)     * DHEAD + mn];
        b0.y = Vp[(size_t)(kc * 4 + kk + 1) * DHEAD + mn];
        b1.x = Vp[(size_t)(kc * 4 + kk)     * DHEAD + 16 + mn];
        b1.y = Vp[(size_t)(kc * 4 + kk + 1) * DHEAD + 16 + mn];
        o0 = __builtin_amdgcn_wmma_f32_16x16x4_f32(
            false, ap, false, b0, (short)0, o0, false, false);
        o1 = __builtin_amdgcn_wmma_f32_16x16x4_f32(
            false, ap, false, b1, (short)0, o1, false, false);
    }
    for (int r = 0; r < 8; ++r) {
        part[(wid * 16 + rowb + r) * 32 + mn]      = o0[r];
        part[(wid * 16 + rowb + r) * 32 + 16 + mn] = o1[r];
    }
    __syncthreads();

    // reduce 4 wave partials, write ctx[b, s, h*32+d] (ready for projection)
    for (int e = threadIdx.x; e < 512; e += 128) {
        const int rr = e >> 5;
        const int dd = e & 31;
        const float v = part[(0 * 16 + rr) * 32 + dd]
                      + part[(1 * 16 + rr) * 32 + dd]
                      + part[(2 * 16 + rr) * 32 + dd]
                      + part[(3 * 16 + rr) * 32 + dd];
        ctx[((size_t)b * SEQ + q0 + rr) * DMODEL + h * DHEAD + dd] = v;
    }
}

// -------------------------------------------------------------------------
// Phase 3: output projection  out = ctx @ wc^T + bc
// Register-blocked 16x64 per wave like phase 1.
// Waves: (4096/16) * (256/64) = 1024.
// -------------------------------------------------------------------------
__global__ __launch_bounds__(128) void proj_kernel(
    const float* __restrict__ ctx,
    const float* __restrict__ wc, const float* __restrict__ bc,
    float* __restrict__ out)
{
    const int wave = (blockIdx.x * blockDim.x + threadIdx.x) >> 5;
    const int lane = threadIdx.x & 31;
    const int m0 = (wave >> 2) << 4;
    const int n0 = (wave & 3) << 6;
    const int mn = lane & 15;
    const int kk = (lane >> 4) << 1;

    v8f acc[4];
    for (int j = 0; j < 4; ++j) {
        const float bb = bc[n0 + j * 16 + mn];
        for (int r = 0; r < 8; ++r) acc[j][r] = bb;
    }

    const float* cr  = ctx + (size_t)(m0 + mn) * DMODEL;
    const float* wr0 = wc + (size_t)(n0 +  0 + mn) * DMODEL;
    const float* wr1 = wc + (size_t)(n0 + 16 + mn) * DMODEL;
    const float* wr2 = wc + (size_t)(n0 + 32 + mn) * DMODEL;
    const float* wr3 = wc + (size_t)(n0 + 48 + mn) * DMODEL;

    for (int kc = 0; kc < DMODEL / 4; ++kc) {
        const int ko = kc * 4 + kk;
        v2f a;  a.x = cr[ko];  a.y = cr[ko + 1];
        v2f b0; b0.x = wr0[ko]; b0.y = wr0[ko + 1];
        v2f b1; b1.x = wr1[ko]; b1.y = wr1[ko + 1];
        v2f b2; b2.x = wr2[ko]; b2.y = wr2[ko + 1];
        v2f b3; b3.x = wr3[ko]; b3.y = wr3[ko + 1];
        acc[0] = __builtin_amdgcn_wmma_f32_16x16x4_f32(
            false, a, false, b0, (short)0, acc[0], false, false);
        acc[1] = __builtin_amdgcn_wmma_f32_16x16x4_f32(
            false, a, false, b1, (short)0, acc[1], false, false);
        acc[2] = __builtin_amdgcn_wmma_f32_16x16x4_f32(
            false, a, false, b2, (short)0, acc[2], false, false);
        acc[3] = __builtin_amdgcn_wmma_f32_16x16x4_f32(
            false, a, false, b3, (short)0, acc[3], false, false);
    }

    const int rowb = m0 + ((lane >> 4) << 3);
    for (int j = 0; j < 4; ++j) {
        for (int r = 0; r < 8; ++r)
            out[(size_t)(rowb + r) * DMODEL + n0 + j * 16 + mn] = acc[j][r];
    }
}

// -------------------------------------------------------------------------
extern "C" void kernel_launch(void* const* d_in, const int* in_sizes, int n_in,
                              void* d_out, int out_size, void* d_ws, size_t ws_size,
                              hipStream_t stream) {
    (void)in_sizes; (void)n_in; (void)out_size; (void)ws_size;

    const float* x  = (const float*)d_in[0];
    const float* wq = (const float*)d_in[1];
    const float* bq = (const float*)d_in[2];
    const float* wk = (const float*)d_in[3];
    const float* bk = (const float*)d_in[4];
    const float* wv = (const float*)d_in[5];
    const float* bv = (const float*)d_in[6];
    const float* wc = (const float*)d_in[7];
    const float* bc = (const float*)d_in[8];

    float* out  = (float*)d_out;                                   // [B,S,D]
    float* attn = out + (size_t)BATCH * SEQ * DMODEL;              // [B,S,S]

    const size_t qkv_elems = (size_t)BATCH * NHEAD * SEQ * DHEAD;  // 1M floats
    float* Q   = (float*)d_ws;
    float* Kt  = Q  + qkv_elems;
    float* V   = Kt + qkv_elems;
    float* ctx = V  + qkv_elems;                                   // [B,S,D]

    // zero the mean-attention region (accumulated via atomics per head)
    hipMemsetAsync(attn, 0, (size_t)BATCH * SEQ * SEQ * sizeof(float), stream);

    // 3072 waves / 4 waves per block
    qkv_kernel<<<768, 128, 0, stream>>>(x, wq, bq, wk, bk, wv, bv, Q, Kt, V);

    // B * H * (S/16) = 2048 workgroups
    attn_kernel<<<2048, 128, 0, stream>>>(Q, Kt, V, ctx, attn);

    // 1024 waves / 4 per block
    proj_kernel<<<256, 128, 0, stream>>>(ctx, wc, bc, out);
}